// InitialContextualNodeModel_49976239456342
// MI455X (gfx1250) — compile-verified
//
#include <hip/hip_runtime.h>

typedef __attribute__((ext_vector_type(2))) float v2f;
typedef __attribute__((ext_vector_type(8))) float v8f;

#define D_EDGE 64
#define WPB 4        // waves per block (wave32)
#define XS 196       // padded LDS row stride (196 % 64 == 4 -> conflict-free lane pattern)

// ---------------------------------------------------------------- zero scratch
__global__ void zero_ws_kernel(float* __restrict__ ws, long total) {
  long i = (long)blockIdx.x * blockDim.x + threadIdx.x;
  if (i < total) ws[i] = 0.0f;
}

// ------------------------------------------------- scatter main edges (fwd/bwd)
__global__ void scatter_dir_kernel(const int* __restrict__ eidx,
                                   const float* __restrict__ eattr,
                                   float* __restrict__ fwdS,
                                   float* __restrict__ bwdS,
                                   float* __restrict__ cntF,
                                   float* __restrict__ cntB, int E) {
  long gid = (long)blockIdx.x * blockDim.x + threadIdx.x;
  long total = (long)E * D_EDGE;
  if (gid >= total) return;
  int e = (int)(gid >> 6);
  int f = (int)(gid & 63);
  int past = eidx[e];       // edge_index[0][e]
  int fut  = eidx[E + e];   // edge_index[1][e]
  float v = eattr[gid];
  atomicAdd(&fwdS[(long)fut  * D_EDGE + f], v);
  atomicAdd(&bwdS[(long)past * D_EDGE + f], v);
  if (f == 0) {
    atomicAdd(&cntF[fut], 1.0f);
    atomicAdd(&cntB[past], 1.0f);
  }
}

// ------------------------------------------------- scatter same-frame edges
__global__ void scatter_frame_kernel(const int* __restrict__ sidx,
                                     const float* __restrict__ sattr,
                                     float* __restrict__ frameS,
                                     float* __restrict__ cntM, int Es) {
  long gid = (long)blockIdx.x * blockDim.x + threadIdx.x;
  long total = (long)Es * D_EDGE;
  if (gid >= total) return;
  int e = (int)(gid >> 6);
  int f = (int)(gid & 63);
  int a = sidx[e];        // early
  int b = sidx[Es + e];   // later
  float v = sattr[gid];
  atomicAdd(&frameS[(long)a * D_EDGE + f], v);
  atomicAdd(&frameS[(long)b * D_EDGE + f], v);
  if (f == 0) {
    atomicAdd(&cntM[a], 1.0f);
    atomicAdd(&cntM[b], 1.0f);
  }
}

// ------------------------------------------------- fused divide + MLP via WMMA
__global__ __launch_bounds__(WPB * 32) void mlp_wmma_kernel(
    const float* __restrict__ fwdS, const float* __restrict__ frameS,
    const float* __restrict__ bwdS, const float* __restrict__ cntF,
    const float* __restrict__ cntM, const float* __restrict__ cntB,
    const float* __restrict__ W1, const float* __restrict__ b1,
    const float* __restrict__ W2, const float* __restrict__ b2,
    float* __restrict__ out, int N) {
  __shared__ float Xs[WPB][16 * XS];   // X tile (16x192) then reused as h tile (16x128)
  __shared__ float inv3[WPB][3][16];

  const int tid = threadIdx.x;
  const int w = tid >> 5;
  const int lane = tid & 31;
  const int rowbase = (blockIdx.x * WPB + w) * 16;

  // per-row inverse counts (rows clamped so all waves stay convergent)
  if (lane < 16) {
    int row = min(rowbase + lane, N - 1);
    inv3[w][0][lane] = 1.0f / fmaxf(cntF[row], 1.0f);
    inv3[w][1][lane] = 1.0f / fmaxf(cntM[row], 1.0f);
    inv3[w][2][lane] = 1.0f / fmaxf(cntB[row], 1.0f);
  }
  __syncthreads();

  // stage X = [fwd_agg | frame_agg | bwd_agg] for 16 rows into LDS
  for (int i = lane; i < 16 * 192; i += 32) {
    int r = i / 192, f = i % 192;
    int row = min(rowbase + r, N - 1);
    float v;
    if (f < 64)       v = fwdS[(long)row * 64 + f]          * inv3[w][0][r];
    else if (f < 128) v = frameS[(long)row * 64 + (f - 64)] * inv3[w][1][r];
    else              v = bwdS[(long)row * 64 + (f - 128)]  * inv3[w][2][r];
    Xs[w][r * XS + f] = v;
  }
  __syncthreads();

  const int m = lane & 15;                 // M-row (A) / N-col (B,C,D) within tile
  const int koff = (lane < 16) ? 0 : 2;    // K sub-offset per A/B f32 layout
  const int mhi = (lane < 16) ? 0 : 8;     // C/D row offset

  // ---- layer 1: [16x192] @ [192x128], 8 output tiles, A reused across tiles
  v8f acc[8] = {};
  for (int k0 = 0; k0 < 192; k0 += 4) {
    v2f a;
    a.x = Xs[w][m * XS + k0 + koff];
    a.y = Xs[w][m * XS + k0 + koff + 1];
#pragma unroll
    for (int nt = 0; nt < 8; ++nt) {
      int col = nt * 16 + m;
      v2f b;
      b.x = W1[(k0 + koff) * 128 + col];
      b.y = W1[(k0 + koff + 1) * 128 + col];
      acc[nt] = __builtin_amdgcn_wmma_f32_16x16x4_f32(
          false, a, false, b, (short)0, acc[nt], false, false);
    }
  }
  __syncthreads();

  // h = relu(acc + b1) -> back into the same LDS region (cols 0..127)
#pragma unroll
  for (int nt = 0; nt < 8; ++nt) {
    int col = nt * 16 + m;
    float bias = b1[col];
#pragma unroll
    for (int r = 0; r < 8; ++r) {
      Xs[w][(r + mhi) * XS + col] = fmaxf(acc[nt][r] + bias, 0.0f);
    }
  }
  __syncthreads();

  // ---- layer 2: [16x128] @ [128x64], 4 output tiles
  v8f acc2[4] = {};
  for (int k0 = 0; k0 < 128; k0 += 4) {
    v2f a;
    a.x = Xs[w][m * XS + k0 + koff];
    a.y = Xs[w][m * XS + k0 + koff + 1];
#pragma unroll
    for (int nt = 0; nt < 4; ++nt) {
      int col = nt * 16 + m;
      v2f b;
      b.x = W2[(k0 + koff) * 64 + col];
      b.y = W2[(k0 + koff + 1) * 64 + col];
      acc2[nt] = __builtin_amdgcn_wmma_f32_16x16x4_f32(
          false, a, false, b, (short)0, acc2[nt], false, false);
    }
  }

  // store out = acc2 + b2 (guard rows)
#pragma unroll
  for (int nt = 0; nt < 4; ++nt) {
    int col = nt * 16 + m;
    float bias = b2[col];
#pragma unroll
    for (int r = 0; r < 8; ++r) {
      int row = rowbase + r + mhi;
      if (row < N) out[(long)row * 64 + col] = acc2[nt][r] + bias;
    }
  }
}

// ---------------------------------------------------------------- launcher
extern "C" void kernel_launch(void* const* d_in, const int* in_sizes, int n_in,
                              void* d_out, int out_size, void* d_ws, size_t ws_size,
                              hipStream_t stream) {
  const int*   eidx  = (const int*)d_in[0];    // edge_index [2,E]
  const float* eattr = (const float*)d_in[1];  // edge_attr  [E,64]
  // d_in[2] = num_nodes (device scalar; N derived from out_size instead)
  const int*   sidx  = (const int*)d_in[3];    // same_frame_edge_index [2,Es]
  const float* sattr = (const float*)d_in[4];  // same_frame_edge_attr  [Es,64]
  const float* W1 = (const float*)d_in[5];     // [192,128]
  const float* b1 = (const float*)d_in[6];     // [128]
  const float* W2 = (const float*)d_in[7];     // [128,64]
  const float* b2 = (const float*)d_in[8];     // [64]
  float* out = (float*)d_out;

  const int E  = in_sizes[0] / 2;
  const int Es = in_sizes[3] / 2;
  const int N  = out_size / 64;

  float* ws     = (float*)d_ws;
  float* fwdS   = ws;
  float* frameS = fwdS + (size_t)N * 64;
  float* bwdS   = frameS + (size_t)N * 64;
  float* cntF   = bwdS + (size_t)N * 64;
  float* cntM   = cntF + N;
  float* cntB   = cntM + N;

  long ztot = (long)N * (3 * 64 + 3);
  zero_ws_kernel<<<(int)((ztot + 255) / 256), 256, 0, stream>>>(ws, ztot);

  long t1 = (long)E * 64;
  scatter_dir_kernel<<<(int)((t1 + 255) / 256), 256, 0, stream>>>(
      eidx, eattr, fwdS, bwdS, cntF, cntB, E);

  long t2 = (long)Es * 64;
  scatter_frame_kernel<<<(int)((t2 + 255) / 256), 256, 0, stream>>>(
      sidx, sattr, frameS, cntM, Es);

  int nblocks = (N + WPB * 16 - 1) / (WPB * 16);
  mlp_wmma_kernel<<<nblocks, WPB * 32, 0, stream>>>(
      fwdS, frameS, bwdS, cntF, cntM, cntB, W1, b1, W2, b2, out, N);
}